// TemporalJitter_67422396613239
// MI455X (gfx1250) — compile-verified
//
#include <hip/hip_runtime.h>
#include <hip/hip_bf16.h>
#include <stdint.h>

#define P_DEL 0.05f
#define B_  16
#define C_  3
#define T_  75
#define H_  112
#define W_  112
#define HW_   (H_ * W_)       // 12544 floats per frame-channel slice
#define HW4_  (HW_ / 4)       // 3136 float4 per slice

// ---------------------------------------------------------------------------
// Kernel 1: per-batch serial scan -> frame source map (faithful to reference).
// out starts as a copy of x, so the map defaults to identity; each conditional
// write claims the next output slot exactly once (cnt is monotone).
// ---------------------------------------------------------------------------
__global__ void jitter_map_kernel(const float* __restrict__ u,
                                  int* __restrict__ map) {
    int b = threadIdx.x;
    if (b >= B_) return;
    const float* ub = u + b * T_;
    int* mb = map + b * T_;
    for (int j = 0; j < T_; ++j) mb[j] = j;          // identity default
    int cnt = 0;
    for (int t = 0; t < T_; ++t) {
        int d  = (ub[t] < P_DEL) ? 1 : 0;            // prob_del[t]
        int dp = (ub[T_ - 1 - t] < P_DEL) ? 1 : 0;   // prob_dup[t] = del reversed
        if (!d && cnt < T_) { mb[cnt] = t; ++cnt; }          // keep frame t
        if (dp && cnt > 0 && cnt < T_) { mb[cnt] = cnt - 1; ++cnt; } // dup INPUT[cnt-1]
    }
}

// ---------------------------------------------------------------------------
// Kernel 2: gather-copy of 50KB frame slices, CDNA5 async global<->LDS path.
// ---------------------------------------------------------------------------
#if defined(__has_builtin)
#  if __has_builtin(__builtin_amdgcn_global_load_async_to_lds_b128) && \
      __has_builtin(__builtin_amdgcn_global_store_async_from_lds_b128)
#    define JITTER_ASYNC 1
#  endif
#endif

// Builtin signature (from clang diagnostics, rounds 1-2):
//   (int4 addrspace(1)* global, int4 addrspace(3)* lds, Ii offset, Ii cpol)
typedef int v4i_t __attribute__((ext_vector_type(4)));
typedef __attribute__((address_space(1))) v4i_t gas_v4i;
typedef __attribute__((address_space(3))) v4i_t las_v4i;

__global__ void __launch_bounds__(256)
jitter_copy_kernel(const float4* __restrict__ x,
                   float4* __restrict__ out,
                   const int* __restrict__ map) {
    const int f  = blockIdx.x;                 // 0 .. B*C*T-1  (one slice per block)
    const int b  = f / (C_ * T_);
    const int ct = f - b * (C_ * T_);
    const int c  = ct / T_;
    const int j  = ct - c * T_;
    const int src = map[b * T_ + j];

    const float4* gsrc = x   + (size_t)((b * C_ + c) * T_ + src) * HW4_;
    float4*       gdst = out + (size_t)f * HW4_;

#ifdef JITTER_ASYNC
    __shared__ float4 buf[HW4_];               // 50176 bytes; 6 blocks/WGP
    // Phase 1: stream global -> LDS, ~13 async ops in flight per lane.
    for (int i = threadIdx.x; i < HW4_; i += 256) {
        __builtin_amdgcn_global_load_async_to_lds_b128(
            (gas_v4i*)(uintptr_t)(gsrc + i),
            (las_v4i*)(uint32_t)(uintptr_t)(buf + i),
            0, 0);
    }
    // Each thread only stores LDS slots it loaded itself -> per-wave wait is
    // sufficient, no barrier needed across waves.
    asm volatile("s_wait_asynccnt 0" ::: "memory");
    // Phase 2: stream LDS -> global.
    for (int i = threadIdx.x; i < HW4_; i += 256) {
        __builtin_amdgcn_global_store_async_from_lds_b128(
            (gas_v4i*)(uintptr_t)(gdst + i),
            (las_v4i*)(uint32_t)(uintptr_t)(buf + i),
            0, 0);
    }
    asm volatile("s_wait_asynccnt 0" ::: "memory");
#else
    // Fallback: plain coalesced B128 copy through VGPRs.
    for (int i = threadIdx.x; i < HW4_; i += 256) {
        gdst[i] = gsrc[i];
    }
#endif
}

// ---------------------------------------------------------------------------
// Launch: inputs are (x, rand_u) per setup_inputs() order; map lives in d_ws.
// ---------------------------------------------------------------------------
extern "C" void kernel_launch(void* const* d_in, const int* in_sizes, int n_in,
                              void* d_out, int out_size, void* d_ws, size_t ws_size,
                              hipStream_t stream) {
    (void)in_sizes; (void)n_in; (void)out_size; (void)ws_size;
    const float* x = (const float*)d_in[0];
    const float* u = (const float*)d_in[1];
    float* out = (float*)d_out;
    int* map = (int*)d_ws;                      // B*T ints = 4800 bytes

    jitter_map_kernel<<<1, 32, 0, stream>>>(u, map);
    jitter_copy_kernel<<<B_ * C_ * T_, 256, 0, stream>>>(
        (const float4*)x, (float4*)out, map);
}